// LIFLayer_30374008718152
// MI455X (gfx1250) — compile-verified
//
#include <hip/hip_runtime.h>

// LIF constants (match reference)
#define LIF_T     512
#define LIF_B     64
#define LIF_N     4096
#define LIF_DECAY 0.05f      // DT/TAU = 1/20
#define LIF_ETA   0.1f
#define LIF_MINTH 0.5f
#define LIF_MAXTH 2.0f
#define LIF_TH0   1.0f

#define CELLS    (LIF_B * LIF_N)   // 262144 floats per timestep (1 MB)
#define VCELLS   (CELLS / 4)       // 65536 float4 groups per timestep
#define DEPTH    8                 // register ring-buffer depth (loads in flight)
#define PF_DIST  16                // prefetch distance in timesteps (16 MB ahead of compute)

typedef __attribute__((ext_vector_type(4))) float v4f;

// One LIF lane step on scalar state held in VGPRs.
__device__ __forceinline__ float lif_step(float I, float& V, float& Th) {
    V = __builtin_fmaf(LIF_DECAY, I - V, V);        // leaky integrate: V += decay*(I-V)
    const bool fire = (V >= Th);                    // heaviside
    const float s = fire ? 1.0f : 0.0f;
    V = fire ? 0.0f : V;                            // reset where spiked
    Th = __builtin_fminf(
             __builtin_fmaxf(__builtin_fmaf(LIF_ETA, s, Th), LIF_MINTH),
             LIF_MAXTH);                            // adaptive threshold, clamped
    return s;
}

__global__ __launch_bounds__(256)
void lif_scan_kernel(const float* __restrict__ in, float* __restrict__ out) {
    const int cell = blockIdx.x * blockDim.x + threadIdx.x;   // 0 .. VCELLS-1

    // Three running pointers, bumped once per 8-step chunk; the 8 accesses
    // inside a chunk use constant ioffsets 0..7 MB (within signed 24-bit range).
    const v4f* __restrict__ pLd = (const v4f*)in  + cell + DEPTH   * VCELLS; // loads t+8
    const v4f* __restrict__ pPf = (const v4f*)in  + cell + PF_DIST * VCELLS; // prefetch t+16
    v4f*       __restrict__ pSt = (v4f*)out       + cell;                    // stores t

    // Recurrence state: 4 neurons per thread.
    float Vm[4] = {0.0f, 0.0f, 0.0f, 0.0f};
    float Th[4] = {LIF_TH0, LIF_TH0, LIF_TH0, LIF_TH0};

    // Prologue: fill the 8-deep ring buffer (t = 0..7) -> 8 NT b128 loads in flight.
    v4f buf[DEPTH];
    {
        const v4f* p0 = (const v4f*)in + cell;
        #pragma unroll
        for (int i = 0; i < DEPTH; ++i)
            buf[i] = __builtin_nontemporal_load(p0 + i * VCELLS);
    }

    auto compute_store = [&](v4f I, v4f* dst) {
        v4f S;
        #pragma unroll
        for (int k = 0; k < 4; ++k) {
            float v = Vm[k], th = Th[k];
            S[k] = lif_step(I[k], v, th);
            Vm[k] = v; Th[k] = th;
        }
        __builtin_nontemporal_store(S, dst);
    };

    // Main loop: 62 chunks x 8 steps = t 0..495.
    // Per step: prefetch t+16 into L2, refill ring slot with load of t+8
    // (wait becomes loadcnt<=DEPTH-1, not 0), compute+store t.
    for (int c = 0; c < (LIF_T - PF_DIST) / DEPTH; ++c) {   // 62 chunks
        #pragma unroll
        for (int i = 0; i < DEPTH; ++i) {
            __builtin_prefetch((const void*)(pPf + i * VCELLS), 0, 0);
            v4f I = buf[i];
            buf[i] = __builtin_nontemporal_load(pLd + i * VCELLS);
            compute_store(I, pSt + i * VCELLS);
        }
        pPf += DEPTH * VCELLS;
        pLd += DEPTH * VCELLS;
        pSt += DEPTH * VCELLS;
    }

    // Tail chunk with loads but no prefetch: compute t=496..503, load t=504..511.
    #pragma unroll
    for (int i = 0; i < DEPTH; ++i) {
        v4f I = buf[i];
        buf[i] = __builtin_nontemporal_load(pLd + i * VCELLS);
        compute_store(I, pSt + i * VCELLS);
    }
    pSt += DEPTH * VCELLS;

    // Drain: compute t=504..511 from the ring buffer, no more loads.
    #pragma unroll
    for (int i = 0; i < DEPTH; ++i)
        compute_store(buf[i], pSt + i * VCELLS);
}

extern "C" void kernel_launch(void* const* d_in, const int* in_sizes, int n_in,
                              void* d_out, int out_size, void* d_ws, size_t ws_size,
                              hipStream_t stream) {
    const float* in  = (const float*)d_in[0];   // input_data [T,B,N] f32
    float*       out = (float*)d_out;           // spikes     [T,B,N] f32

    dim3 block(256);                 // 8 wave32 waves per block
    dim3 grid(VCELLS / 256);         // 65536 threads = one thread per 4 neurons
    lif_scan_kernel<<<grid, block, 0, stream>>>(in, out);
}